// QLSTM_65481071410705
// MI455X (gfx1250) — compile-verified
//
#include <hip/hip_runtime.h>
#include <math.h>

typedef __attribute__((ext_vector_type(2))) float v2f;
typedef __attribute__((ext_vector_type(8))) float v8f;

#define NQ     8
#define DIM    128
#define BATCH  128
#define TSTEPS 64
#define KDIM   (DIM + NQ)   // 136
#define BB     2            // batch elements per block (8 waves -> 1 circuit/wave)
#define CSTR   137          // comb row stride (floats)
#define WSTR   33           // transposed-weight row stride

struct c32 { float x, y; };

__device__ __forceinline__ float fast_sigmoid(float x) {
    return 1.f / (1.f + __expf(-x));
}
__device__ __forceinline__ float fast_tanh(float x) {
    float e2 = __expf(2.f * x);
    return 1.f - 2.f / (e2 + 1.f);
}

// State layout: amp index = lane*8 + r  (lane -> bits [7:3], r -> bits [2:0]).

// ---- RX(theta): [[c,-is],[-is,c]] — symmetric, no select needed cross-lane
template<int BBIT>
__device__ __forceinline__ void rx_apply(c32* st, float cc, float ss, int lane) {
    if constexpr (BBIT < 3) {
        constexpr int STP = 1 << BBIT;
        #pragma unroll
        for (int base = 0; base < 8; ++base) {
            if ((base & STP) == 0) {
                c32 a0 = st[base], a1 = st[base + STP];
                st[base]       = c32{ cc * a0.x + ss * a1.y, cc * a0.y - ss * a1.x };
                st[base + STP] = c32{ ss * a0.y + cc * a1.x, cc * a1.y - ss * a0.x };
            }
        }
    } else {
        constexpr int MASK = 1 << (BBIT - 3);
        #pragma unroll
        for (int r = 0; r < 8; ++r) {
            c32 mine = st[r];
            c32 oth;
            oth.x = __shfl_xor(mine.x, MASK, 32);
            oth.y = __shfl_xor(mine.y, MASK, 32);
            st[r] = c32{ cc * mine.x + ss * oth.y, cc * mine.y - ss * oth.x };
        }
    }
    (void)lane;
}

// ---- RY(theta): [[c,-s],[s,c]] — one per-lane sign cross-lane
template<int BBIT>
__device__ __forceinline__ void ry_apply(c32* st, float cc, float ss, int lane) {
    if constexpr (BBIT < 3) {
        constexpr int STP = 1 << BBIT;
        #pragma unroll
        for (int base = 0; base < 8; ++base) {
            if ((base & STP) == 0) {
                c32 a0 = st[base], a1 = st[base + STP];
                st[base]       = c32{ cc * a0.x - ss * a1.x, cc * a0.y - ss * a1.y };
                st[base + STP] = c32{ ss * a0.x + cc * a1.x, ss * a0.y + cc * a1.y };
            }
        }
    } else {
        constexpr int MASK = 1 << (BBIT - 3);
        const float sg = (lane & MASK) ? ss : -ss;
        #pragma unroll
        for (int r = 0; r < 8; ++r) {
            c32 mine = st[r];
            c32 oth;
            oth.x = __shfl_xor(mine.x, MASK, 32);
            oth.y = __shfl_xor(mine.y, MASK, 32);
            st[r] = c32{ cc * mine.x + sg * oth.x, cc * mine.y + sg * oth.y };
        }
    }
}

// ---- RZ(theta): diag(e^{-ih}, e^{+ih}) — pure phase, never shuffles
template<int BBIT>
__device__ __forceinline__ void rz_apply(c32* st, float cc, float ss, int lane) {
    if constexpr (BBIT < 3) {
        #pragma unroll
        for (int r = 0; r < 8; ++r) {
            const float sg = (r & (1 << BBIT)) ? ss : -ss;   // compile-time sign
            c32 a = st[r];
            st[r] = c32{ cc * a.x - sg * a.y, cc * a.y + sg * a.x };
        }
    } else {
        constexpr int MASK = 1 << (BBIT - 3);
        const float sg = (lane & MASK) ? ss : -ss;
        #pragma unroll
        for (int r = 0; r < 8; ++r) {
            c32 a = st[r];
            st[r] = c32{ cc * a.x - sg * a.y, cc * a.y + sg * a.x };
        }
    }
}

// encoder gate on wire Q (family cycles rx/ry/rz), fast hw trig
template<int Q>
__device__ __forceinline__ void enc_gate(c32* st, float theta, int lane) {
    constexpr int BBIT = 7 - Q;
    float hh = 0.5f * theta;
    float cc = __cosf(hh), ss = __sinf(hh);
    if constexpr (Q % 3 == 0)      rx_apply<BBIT>(st, cc, ss, lane);
    else if constexpr (Q % 3 == 1) ry_apply<BBIT>(st, cc, ss, lane);
    else                           rz_apply<BBIT>(st, cc, ss, lane);
}

// Full adjacent-CNOT chain (ctrl 0..6) == Gray-code gather: src = idx ^ (idx>>1)
__device__ __forceinline__ void cnot_chain(c32* st, int lane) {
    const int srcLane = (lane ^ (lane >> 1)) & 31;
    c32 t[8];
    #pragma unroll
    for (int r = 0; r < 8; ++r) {
        t[r].x = __shfl(st[r].x, srcLane, 32);
        t[r].y = __shfl(st[r].y, srcLane, 32);
    }
    const bool odd = (lane & 1) != 0;   // idx bit3 = lane bit0 folds into reg bit2
    #pragma unroll
    for (int r = 0; r < 8; ++r) {
        const int g = (r ^ (r >> 1)) & 7;        // compile-time Gray remap
        c32 a = t[g], b = t[g ^ 4];
        st[r].x = odd ? b.x : a.x;
        st[r].y = odd ? b.y : a.y;
    }
}

__device__ __forceinline__ void measure_z(const c32* st, float* outRow, int lane) {
    float p[8];
    #pragma unroll
    for (int r = 0; r < 8; ++r)
        p[r] = st[r].x * st[r].x + st[r].y * st[r].y;

    // 3-bit Hadamard combine over the register bits
    float t01 = p[0] + p[1], t23 = p[2] + p[3], t45 = p[4] + p[5], t67 = p[6] + p[7];
    float u03 = t01 + t23, u47 = t45 + t67;
    float plane = u03 + u47;

    float ex[8];
    ex[5] = u03 - u47;                                   // reg bit2 sign (wire 5)
    ex[6] = (t01 + t45) - (t23 + t67);                   // reg bit1 sign (wire 6)
    ex[7] = (p[0] + p[2] + p[4] + p[6]) -
            (p[1] + p[3] + p[5] + p[7]);                 // reg bit0 sign (wire 7)
    #pragma unroll
    for (int w = 0; w < 5; ++w) {                        // wires 0..4: lane-bit signs
        const int lb = 4 - w;
        ex[w] = ((lane >> lb) & 1) ? -plane : plane;
    }

    #pragma unroll
    for (int w = 0; w < 8; ++w) {
        #pragma unroll
        for (int off = 16; off > 0; off >>= 1)
            ex[w] += __shfl_xor(ex[w], off, 32);
    }
    if (lane == 0) {
        #pragma unroll
        for (int w = 0; w < 8; ++w) outRow[w] = ex[w];
    }
}

__global__ __launch_bounds__(256) void qlstm_persistent(
    const float* __restrict__ inputs,
    const float* __restrict__ Wf, const float* __restrict__ bf,
    const float* __restrict__ Wi, const float* __restrict__ bi,
    const float* __restrict__ Wu, const float* __restrict__ bu,
    const float* __restrict__ Wo, const float* __restrict__ bo,
    const float* __restrict__ pf, const float* __restrict__ pi,
    const float* __restrict__ pu, const float* __restrict__ po,
    float* __restrict__ ys, float* __restrict__ hx_out, float* __restrict__ cx_out)
{
    __shared__ float sWt[KDIM][WSTR];      // transposed fused weights [k][g*8+o]
    __shared__ float sB[32];
    __shared__ float sPc[32], sPs[32];     // cos/sin of trainable RX half-angles
    __shared__ float sComb[16][CSTR];      // rows BB..15 stay zero (WMMA pad)
    __shared__ float sAng[BB][32];
    __shared__ float sExp[4][BB][NQ];
    __shared__ float sH[BB][NQ], sC[BB][NQ];

    const int tid  = threadIdx.x;
    const int lane = tid & 31;
    const int wv   = tid >> 5;             // 8 waves
    const int b0   = blockIdx.x * BB;

    // ---- one-time setup -------------------------------------------------
    for (int e = tid; e < NQ * KDIM; e += 256) {
        int o = e / KDIM, k = e - o * KDIM;
        sWt[k][ 0 + o] = Wf[e];
        sWt[k][ 8 + o] = Wi[e];
        sWt[k][16 + o] = Wu[e];
        sWt[k][24 + o] = Wo[e];
    }
    if (tid < 8) {
        sB[tid] = bf[tid]; sB[8 + tid] = bi[tid]; sB[16 + tid] = bu[tid]; sB[24 + tid] = bo[tid];
        float a0 = 0.5f * pf[tid], a1 = 0.5f * pi[tid];
        float a2 = 0.5f * pu[tid], a3 = 0.5f * po[tid];
        sPc[tid]      = __cosf(a0); sPs[tid]      = __sinf(a0);
        sPc[8 + tid]  = __cosf(a1); sPs[8 + tid]  = __sinf(a1);
        sPc[16 + tid] = __cosf(a2); sPs[16 + tid] = __sinf(a2);
        sPc[24 + tid] = __cosf(a3); sPs[24 + tid] = __sinf(a3);
    }
    for (int e = tid; e < (16 - BB) * CSTR; e += 256)    // zero WMMA pad rows
        sComb[BB + e / CSTR][e % CSTR] = 0.f;
    if (tid < BB * NQ) { sH[tid / NQ][tid % NQ] = 0.f; sC[tid / NQ][tid % NQ] = 0.f; }
    __syncthreads();

    // ---- recurrence -----------------------------------------------------
    for (int t = 0; t < TSTEPS; ++t) {
        // phase 1: comb = [x_t, h]
        for (int e = tid; e < BB * KDIM; e += 256) {
            int j = e / KDIM, k = e - j * KDIM;
            float v = (k < DIM)
                ? inputs[(size_t)t * BATCH * DIM + (size_t)(b0 + j) * DIM + k]
                : sH[j][k - DIM];
            sComb[j][k] = v;
        }
        __syncthreads();

        // phase 2: fused GEMM [16x136]x[136x32] via V_WMMA_F32_16X16X4_F32
        if (wv < 2) {                      // wave 0 -> N tile 0..15, wave 1 -> 16..31
            const int m = lane & 15;
            const int h = lane >> 4;
            const int n = (lane & 15) + wv * 16;
            v8f acc = {};
            for (int kk = 0; kk < KDIM; kk += 4) {
                v2f a, b;
                a[0] = sComb[m][kk + 2 * h];
                a[1] = sComb[m][kk + 2 * h + 1];
                b[0] = sWt[kk + 2 * h][n];
                b[1] = sWt[kk + 2 * h + 1][n];
                acc = __builtin_amdgcn_wmma_f32_16x16x4_f32(
                        false, a, false, b, (short)0, acc, false, false);
            }
            #pragma unroll
            for (int r = 0; r < 8; ++r) {
                int mm = r + 8 * h;        // C/D layout: M = vgpr + 8*(lane>=16)
                if (mm < BB) sAng[mm][n] = acc[r] + sB[n];
            }
        }
        __syncthreads();

        // phase 3: statevector sim; wave wv owns batch j=wv>>2, gate g=wv&3
        {
            const int j    = wv >> 2;
            const int gb   = (wv & 3) * NQ;        // base into fused 32-wide arrays
            c32 st[8];
            #pragma unroll
            for (int r = 0; r < 8; ++r) {
                float v0 = (lane == 0 && r == 0) ? 1.f : 0.f;
                st[r] = c32{v0, 0.f};
            }

            #define APPLY_ENC(Q)  enc_gate<Q>(st, sAng[j][gb + Q], lane);
            APPLY_ENC(0) APPLY_ENC(1) APPLY_ENC(2) APPLY_ENC(3)
            APPLY_ENC(4) APPLY_ENC(5) APPLY_ENC(6) APPLY_ENC(7)
            #undef APPLY_ENC

            #define APPLY_RX(Q)   rx_apply<7 - Q>(st, sPc[gb + Q], sPs[gb + Q], lane);
            APPLY_RX(0) APPLY_RX(1) APPLY_RX(2) APPLY_RX(3)
            APPLY_RX(4) APPLY_RX(5) APPLY_RX(6) APPLY_RX(7)
            #undef APPLY_RX

            cnot_chain(st, lane);

            measure_z(st, &sExp[wv & 3][j][0], lane);
        }
        __syncthreads();

        // phase 4: LSTM cell update
        if (tid < BB * NQ) {
            int j = tid / NQ, k = tid % NQ;
            float f  = fast_sigmoid(sExp[0][j][k]);
            float ii = fast_sigmoid(sExp[1][j][k]);
            float gg = fast_tanh(sExp[2][j][k]);
            float oo = fast_sigmoid(sExp[3][j][k]);
            float cn = f * sC[j][k] + ii * gg;
            float hn = oo * fast_tanh(cn);
            sC[j][k] = cn;
            sH[j][k] = hn;
            ys[(size_t)t * BATCH * NQ + (size_t)(b0 + j) * NQ + k] = hn;
        }
        __syncthreads();
    }

    if (tid < BB * NQ) {
        int j = tid / NQ, k = tid % NQ;
        hx_out[(size_t)(b0 + j) * NQ + k] = sH[j][k];
        cx_out[(size_t)(b0 + j) * NQ + k] = sC[j][k];
    }
}

extern "C" void kernel_launch(void* const* d_in, const int* in_sizes, int n_in,
                              void* d_out, int out_size, void* d_ws, size_t ws_size,
                              hipStream_t stream) {
    (void)in_sizes; (void)n_in; (void)out_size; (void)d_ws; (void)ws_size;
    const float* inputs = (const float*)d_in[0];
    const float* Wf = (const float*)d_in[1];
    const float* bf = (const float*)d_in[2];
    const float* Wi = (const float*)d_in[3];
    const float* bi = (const float*)d_in[4];
    const float* Wu = (const float*)d_in[5];
    const float* bu = (const float*)d_in[6];
    const float* Wo = (const float*)d_in[7];
    const float* bo = (const float*)d_in[8];
    const float* pf = (const float*)d_in[9];
    const float* pi = (const float*)d_in[10];
    const float* pu = (const float*)d_in[11];
    const float* po = (const float*)d_in[12];

    float* ys = (float*)d_out;
    float* hx = ys + (size_t)TSTEPS * BATCH * NQ;   // 65536
    float* cx = hx + (size_t)BATCH * NQ;            // +1024

    qlstm_persistent<<<BATCH / BB, 256, 0, stream>>>(
        inputs, Wf, bf, Wi, bi, Wu, bu, Wo, bo, pf, pi, pu, po, ys, hx, cx);
}